// SymbiosGNN_48971217109554
// MI455X (gfx1250) — compile-verified
//
#include <hip/hip_runtime.h>

// GCN 2-layer forward for MI455X (gfx1250, wave32).
// Memory-bound: ~1.2 GB of edge gather/scatter traffic dominates; GEMMs are
// ~185 MFLOP total and done with V_WMMA_F32_16X16X4_F32 per-wave 16-row tiles.

#define NNODES 100000
#define NEDGES 3200000
#define INCH   13
#define HIDCH  32
#define LATCH  16

typedef __attribute__((ext_vector_type(2))) float v2f;
typedef __attribute__((ext_vector_type(8))) float v8f;

// ---------------- utility ----------------

__global__ void gcn_zero_kernel(float4* __restrict__ p, int n4) {
    int i = blockIdx.x * blockDim.x + threadIdx.x;
    if (i < n4) p[i] = make_float4(0.f, 0.f, 0.f, 0.f);
}

// deg[d] += 1 over real edges (self-loop +1 folded into rsqrt kernel)
__global__ void gcn_degree_kernel(const int* __restrict__ dst, float* __restrict__ deg) {
    int e = blockIdx.x * blockDim.x + threadIdx.x;
    if (e < NEDGES) atomicAdd(&deg[dst[e]], 1.0f);
}

// dis[i] = 1/sqrt(deg[i] + 1)   (deg includes self-loop => always > 0)
__global__ void gcn_rsqrt_kernel(float* __restrict__ dis) {
    int i = blockIdx.x * blockDim.x + threadIdx.x;
    if (i < NNODES) dis[i] = rsqrtf(dis[i] + 1.0f);
}

// ---------------- layer 1 GEMM: pre1[i,c] = (x[i,:] @ W1)[c] * dis[i] ----------------
// One wave handles a 16-row tile; K padded 13->16 (4 wmma steps), N=32 (2 tiles).
// A layout (16x4 f32): lanes 0-15 hold (K=0,K=1), lanes 16-31 hold (K=2,K=3), M = lane&15.
// B/C/D: row striped across lanes within a VGPR.
__global__ void gcn_gemm1_wmma(const float* __restrict__ x, const float* __restrict__ W1,
                               const float* __restrict__ dis, float* __restrict__ pre1) {
    const int lane = threadIdx.x & 31;
    const int wave = threadIdx.x >> 5;
    const int tile = blockIdx.x * (blockDim.x >> 5) + wave;
    const int ntiles = NNODES / 16;   // 6250, exact
    if (tile >= ntiles) return;       // wave-uniform: EXEC stays all-1s for WMMA
    const int row0 = tile * 16;
    const int half = lane >> 4;       // 0: K pair {0,1}; 1: K pair {2,3} of each step
    const int l    = lane & 15;
    const int r    = row0 + l;        // A-matrix row owned by this lane

    v8f c0 = {};  // cols 0..15
    v8f c1 = {};  // cols 16..31
#pragma unroll
    for (int kk = 0; kk < 4; ++kk) {
        const int k0 = kk * 4 + half * 2;
        v2f a, b0, b1;
        a[0]  = (k0     < INCH) ? x[r * INCH + k0]     : 0.0f;
        a[1]  = (k0 + 1 < INCH) ? x[r * INCH + k0 + 1] : 0.0f;
        b0[0] = (k0     < INCH) ? W1[(k0)     * HIDCH + l] : 0.0f;
        b0[1] = (k0 + 1 < INCH) ? W1[(k0 + 1) * HIDCH + l] : 0.0f;
        b1[0] = (k0     < INCH) ? W1[(k0)     * HIDCH + 16 + l] : 0.0f;
        b1[1] = (k0 + 1 < INCH) ? W1[(k0 + 1) * HIDCH + 16 + l] : 0.0f;
        c0 = __builtin_amdgcn_wmma_f32_16x16x4_f32(false, a, false, b0, (short)0, c0, false, false);
        c1 = __builtin_amdgcn_wmma_f32_16x16x4_f32(false, a, false, b1, (short)0, c1, false, false);
    }
#pragma unroll
    for (int v = 0; v < 8; ++v) {
        const int m = row0 + half * 8 + v;   // C/D: VGPR v holds row v (lanes<16) / 8+v
        const float s = dis[m];
        pre1[m * HIDCH + l]      = c0[v] * s;
        pre1[m * HIDCH + 16 + l] = c1[v] * s;
    }
}

// ---------------- edge scatter, 32 channels: acc1[dst] += pre1[src] ----------------
// 8 threads per edge, float4 per thread: a wave covers 4 edges => 4 cache lines
// per vector load/atomic instead of 32.
__global__ void gcn_scatter1_kernel(const int* __restrict__ src, const int* __restrict__ dst,
                                    const float* __restrict__ pre1, float* __restrict__ acc1) {
    int gid = blockIdx.x * blockDim.x + threadIdx.x;   // < E*8 = 25.6M
    int e  = gid >> 3;
    int cg = (gid & 7) * 4;
    if (e >= NEDGES) return;
    int s = src[e], d = dst[e];
    const float4 m = *(const float4*)(pre1 + s * HIDCH + cg);
    float* out = acc1 + d * HIDCH + cg;
    atomicAdd(out + 0, m.x);
    atomicAdd(out + 1, m.y);
    atomicAdd(out + 2, m.z);
    atomicAdd(out + 3, m.w);
}

// ---------------- fused layer-1 finalize + layer-2 GEMM ----------------
// t[i,k] = relu(dis[i]*(acc1[i,k] + pre1[i,k]) + b1[k])   (self-loop = +pre1)
// pre2[i,c] = (t[i,:] @ W2)[c] * dis[i]
// A operand built on the fly per K-step; K=32 -> 8 wmma steps; N=16 -> 1 tile.
__global__ void gcn_gemm2_wmma(const float* __restrict__ pre1, const float* __restrict__ acc1,
                               const float* __restrict__ W2, const float* __restrict__ b1,
                               const float* __restrict__ dis, float* __restrict__ pre2) {
    const int lane = threadIdx.x & 31;
    const int wave = threadIdx.x >> 5;
    const int tile = blockIdx.x * (blockDim.x >> 5) + wave;
    const int ntiles = NNODES / 16;
    if (tile >= ntiles) return;
    const int row0 = tile * 16;
    const int half = lane >> 4;
    const int l    = lane & 15;
    const int r    = row0 + l;
    const float dr = dis[r];

    v8f c = {};
#pragma unroll
    for (int kk = 0; kk < 8; ++kk) {
        const int k0 = kk * 4 + half * 2;
        v2f a, b;
        float t0 = dr * (acc1[r * HIDCH + k0]     + pre1[r * HIDCH + k0])     + b1[k0];
        float t1 = dr * (acc1[r * HIDCH + k0 + 1] + pre1[r * HIDCH + k0 + 1]) + b1[k0 + 1];
        a[0] = fmaxf(t0, 0.0f);
        a[1] = fmaxf(t1, 0.0f);
        b[0] = W2[(k0)     * LATCH + l];
        b[1] = W2[(k0 + 1) * LATCH + l];
        c = __builtin_amdgcn_wmma_f32_16x16x4_f32(false, a, false, b, (short)0, c, false, false);
    }
#pragma unroll
    for (int v = 0; v < 8; ++v) {
        const int m = row0 + half * 8 + v;
        pre2[m * LATCH + l] = c[v] * dis[m];
    }
}

// ---------------- edge scatter, 16 channels ----------------
__global__ void gcn_scatter2_kernel(const int* __restrict__ src, const int* __restrict__ dst,
                                    const float* __restrict__ pre2, float* __restrict__ acc2) {
    int gid = blockIdx.x * blockDim.x + threadIdx.x;   // < E*4 = 12.8M
    int e  = gid >> 2;
    int cg = (gid & 3) * 4;
    if (e >= NEDGES) return;
    int s = src[e], d = dst[e];
    const float4 m = *(const float4*)(pre2 + s * LATCH + cg);
    float* out = acc2 + d * LATCH + cg;
    atomicAdd(out + 0, m.x);
    atomicAdd(out + 1, m.y);
    atomicAdd(out + 2, m.z);
    atomicAdd(out + 3, m.w);
}

// ---------------- finalize: z = dis*(acc2 + pre2) + b2 ----------------
__global__ void gcn_finalize_kernel(const float* __restrict__ pre2, const float* __restrict__ acc2,
                                    const float* __restrict__ dis, const float* __restrict__ b2,
                                    float* __restrict__ out) {
    int gid = blockIdx.x * blockDim.x + threadIdx.x;   // N*4 threads, float4 each
    if (gid >= NNODES * 4) return;
    int node = gid >> 2;
    int cg   = (gid & 3) * 4;
    float s = dis[node];
    const float4 p = *(const float4*)(pre2 + node * LATCH + cg);
    const float4 a = *(const float4*)(acc2 + node * LATCH + cg);
    float4 r;
    r.x = s * (p.x + a.x) + b2[cg + 0];
    r.y = s * (p.y + a.y) + b2[cg + 1];
    r.z = s * (p.z + a.z) + b2[cg + 2];
    r.w = s * (p.w + a.w) + b2[cg + 3];
    *(float4*)(out + node * LATCH + cg) = r;
}

extern "C" void kernel_launch(void* const* d_in, const int* in_sizes, int n_in,
                              void* d_out, int out_size, void* d_ws, size_t ws_size,
                              hipStream_t stream) {
    (void)in_sizes; (void)n_in; (void)out_size; (void)ws_size;
    const float* x   = (const float*)d_in[0];
    const int*   ei  = (const int*)d_in[1];        // [2, E] int32 (JAX x64-off)
    const float* W1  = (const float*)d_in[2];      // [13,32]
    const float* b1  = (const float*)d_in[3];      // [32]
    const float* W2  = (const float*)d_in[4];      // [32,16]
    const float* b2  = (const float*)d_in[5];      // [16]
    const int* src = ei;            // edge_index[0]
    const int* dst = ei + NEDGES;   // edge_index[1]
    float* z = (float*)d_out;

    // workspace layout (floats): dis | pre1 (32N) | acc1 (32N) | pre2 (16N) | acc2 (16N)
    float* ws   = (float*)d_ws;
    float* dis  = ws;
    float* pre1 = dis  + NNODES;
    float* acc1 = pre1 + NNODES * HIDCH;
    float* pre2 = acc1 + NNODES * HIDCH;
    float* acc2 = pre2 + NNODES * LATCH;
    const int total_ws_floats = NNODES * (1 + 2 * HIDCH + 2 * LATCH);  // 9.7M
    const int n4 = total_ws_floats / 4;

    const int B = 256;
    gcn_zero_kernel<<<(n4 + B - 1) / B, B, 0, stream>>>((float4*)d_ws, n4);
    gcn_degree_kernel<<<(NEDGES + B - 1) / B, B, 0, stream>>>(dst, dis);
    gcn_rsqrt_kernel<<<(NNODES + B - 1) / B, B, 0, stream>>>(dis);

    const int ntiles = NNODES / 16;                 // 6250 wave-tiles
    const int gemm_blocks = (ntiles + 7) / 8;       // 8 waves per 256-thread block
    gcn_gemm1_wmma<<<gemm_blocks, B, 0, stream>>>(x, W1, dis, pre1);
    gcn_scatter1_kernel<<<(NEDGES * 8 + B - 1) / B, B, 0, stream>>>(src, dst, pre1, acc1);
    gcn_gemm2_wmma<<<gemm_blocks, B, 0, stream>>>(pre1, acc1, W2, b1, dis, pre2);
    gcn_scatter2_kernel<<<(NEDGES * 4 + B - 1) / B, B, 0, stream>>>(src, dst, pre2, acc2);
    gcn_finalize_kernel<<<(NNODES * 4 + B - 1) / B, B, 0, stream>>>(pre2, acc2, dis, b2, z);
}